// _PLHEACircuit_47175920779402
// MI455X (gfx1250) — compile-verified
//
#include <hip/hip_runtime.h>

#define NW 14
#define DIM 16384
#define NTHREADS 1024
#define NWAVES 32
#define NCOLBLK 128   // DIM/8 octet-columns / 16 cols per wmma block
#define NBLKIT (NCOLBLK / NWAVES)   // 4: compile-time trip count per wave

typedef __attribute__((ext_vector_type(2))) float v2f;
typedef __attribute__((ext_vector_type(8))) float v8f;

// Inverse of the 14-CNOT ring permutation (control (i+1)%14 -> target i, i=0..13)
// acting on basis-state indices (wire 0 = MSB).  src = T^{-1}(k):
//   s_q = parity(k & 0x1FFF & ((1<<(q+1))-1)) ^ k_13 (q<=12),  s_13 = parity(k & 0x1FFF)
__device__ __forceinline__ int perm_inv(int k) {
  int t = k & 0x1FFF;
  t ^= t << 1;
  t ^= t << 2;
  t ^= t << 4;
  t ^= t << 8;           // prefix-XOR: bit q = parity of bits 0..q
  t &= 0x3FFF;
  return (k & 0x2000) ? (t ^ 0x1FFF) : t;
}

// One WMMA gate pass over the full state: D = M16 * X for 2048 octet-columns.
// PERM folds the pending CNOT-ring permutation into the source addressing.
template <bool PERM>
__device__ __forceinline__ void gate_pass(const float* __restrict__ reS,
                                          const float* __restrict__ imS,
                                          float* __restrict__ reD,
                                          float* __restrict__ imD,
                                          const float (*M16)[16],
                                          int p0, int lane, int wave) {
  const int m = lane & 15;
  const int kb = (lane >> 4) << 1;           // 0 (lanes 0-15) / 2 (lanes 16-31)
  // A fragments: chunk c = columns 4c..4c+3 of M16 (ISA 16x4 f32 layout)
  v2f a0, a1, a2, a3;
  a0[0] = M16[m][kb];      a0[1] = M16[m][kb + 1];
  a1[0] = M16[m][kb + 4];  a1[1] = M16[m][kb + 5];
  a2[0] = M16[m][kb + 8];  a2[1] = M16[m][kb + 9];
  a3[0] = M16[m][kb + 12]; a3[1] = M16[m][kb + 13];
  const int lowmask = (1 << p0) - 1;
  #pragma unroll 1
  for (int it = 0; it < NBLKIT; ++it) {      // fixed trip count: no exec masking,
    int cb = wave + it * NWAVES;             // EXEC stays all-ones around WMMA
    int col = cb * 16 + m;
    int base = ((col & ~lowmask) << 3) | (col & lowmask);
    int i0 = base | (kb << p0);
    int i1 = base | ((kb + 1) << p0);
    int i2 = base | ((kb + 4) << p0);
    int i3 = base | ((kb + 5) << p0);
    if (PERM) {
      i0 = perm_inv(i0); i1 = perm_inv(i1);
      i2 = perm_inv(i2); i3 = perm_inv(i3);
    }
    // B fragments (4x16 f32 layout): rows kb,kb+1 per K-chunk
    v2f b0, b1, b2, b3;
    b0[0] = reS[i0]; b0[1] = reS[i1];        // chunk0: X rows 0..3 (re 0..3)
    b1[0] = reS[i2]; b1[1] = reS[i3];        // chunk1: X rows 4..7 (re 4..7)
    b2[0] = imS[i0]; b2[1] = imS[i1];        // chunk2: X rows 8..11 (im 0..3)
    b3[0] = imS[i2]; b3[1] = imS[i3];        // chunk3: X rows 12..15 (im 4..7)
    v8f acc = {};
    acc = __builtin_amdgcn_wmma_f32_16x16x4_f32(false, a0, false, b0, (short)0, acc, false, false);
    acc = __builtin_amdgcn_wmma_f32_16x16x4_f32(false, a1, false, b1, (short)0, acc, false, false);
    acc = __builtin_amdgcn_wmma_f32_16x16x4_f32(false, a2, false, b2, (short)0, acc, false, false);
    acc = __builtin_amdgcn_wmma_f32_16x16x4_f32(false, a3, false, b3, (short)0, acc, false, false);
    // D rows v (lanes 0-15 -> re) / v+8 (lanes 16-31 -> im); stores at natural indices
    float* dst = (lane < 16) ? reD : imD;
    #pragma unroll
    for (int v = 0; v < 8; ++v) dst[base | ((v) << p0)] = acc[v];
  }
}

__global__ __launch_bounds__(NTHREADS, 1)
void qcirc_kernel(const float* __restrict__ x, const float* __restrict__ wts,
                  float* __restrict__ out) {
  __shared__ float sRe[2][DIM];     // ping-pong state, real parts   (128 KB)
  __shared__ float sIm[2][DIM];     // ping-pong state, imag parts   (128 KB)
  __shared__ float M16[16][16];     // real embedding of 3-qubit gate (1 KB)
  __shared__ float Ur[15][2][2];    // per-wire composed 2x2, [14] = identity
  __shared__ float Ui[15][2][2];
  __shared__ float c14[NW], s14[NW];
  __shared__ float wsum[NWAVES];

  const int tid = threadIdx.x;
  const int lane = tid & 31;
  const int wave = tid >> 5;
  const int sample = blockIdx.x;
  const float* xs = x + sample * 28;

  // ---- encode block 1: RX on each wire applied to |0..0> = product state ----
  if (tid < NW) {
    float h = 0.5f * xs[tid];
    c14[tid] = cosf(h);
    s14[tid] = sinf(h);
  }
  if (tid == 14) {
    Ur[14][0][0] = 1.f; Ur[14][0][1] = 0.f; Ur[14][1][0] = 0.f; Ur[14][1][1] = 1.f;
    Ui[14][0][0] = 0.f; Ui[14][0][1] = 0.f; Ui[14][1][0] = 0.f; Ui[14][1][1] = 0.f;
  }
  __syncthreads();
  for (int k = tid; k < DIM; k += NTHREADS) {
    float mag = 1.f;
    #pragma unroll
    for (int p = 0; p < NW; ++p)               // index bit p <-> wire 13-p
      mag *= ((k >> p) & 1) ? s14[13 - p] : c14[13 - p];
    int pc = __popc(k) & 3;                    // phase (-i)^popcount
    float re = 0.f, im = 0.f;
    if (pc == 0) re = mag;
    else if (pc == 1) im = -mag;
    else if (pc == 2) re = -mag;
    else im = mag;
    sRe[0][k] = re;
    sIm[0][k] = im;
  }

  int cur = 0;

  static const int p0tab[5] = {11, 8, 5, 2, 0};
  static const signed char wtab[5][3] = {{0,1,2},{3,4,5},{6,7,8},{9,10,11},{14,12,13}};
  // segment = {layer0, layer1, encode2, layer2, layer3}; first pass of a segment
  // consumes the pending CNOT-ring permutation from the previous layer (if any)
  static const bool segperm[5] = {false, true, true, false, true};

  #pragma unroll 1
  for (int seg = 0; seg < 5; ++seg) {
    __syncthreads();
    if (tid < NW) {
      if (seg == 2) {                          // encode block 2: RX(x[:,14+i]) on wire i
        float h = 0.5f * xs[14 + tid];
        float c = cosf(h), s = sinf(h);
        Ur[tid][0][0] = c;   Ur[tid][0][1] = 0.f;
        Ur[tid][1][0] = 0.f; Ur[tid][1][1] = c;
        Ui[tid][0][0] = 0.f; Ui[tid][0][1] = -s;
        Ui[tid][1][0] = -s;  Ui[tid][1][1] = 0.f;
      } else {                                 // U_i = RY(w2)*RZ(w1)*RY(w0)
        int l = (seg < 2) ? seg : seg - 1;
        const float* wl = wts + l * 3 * NW;
        float a = 0.5f * wl[0 * NW + tid];
        float b = 0.5f * wl[1 * NW + tid];
        float c = 0.5f * wl[2 * NW + tid];
        float ca = cosf(a), sa = sinf(a);
        float cb = cosf(b), sb = sinf(b);
        float cc = cosf(c), sc = sinf(c);
        float cca = cc * ca, ssa = sc * sa, csa = cc * sa, sca = sc * ca;
        Ur[tid][0][0] =  cb * (cca - ssa);  Ui[tid][0][0] = -sb * (cca + ssa);
        Ur[tid][0][1] = -cb * (csa + sca);  Ui[tid][0][1] =  sb * (csa - sca);
        Ur[tid][1][0] =  cb * (sca + csa);  Ui[tid][1][0] =  sb * (csa - sca);
        Ur[tid][1][1] =  cb * (cca - ssa);  Ui[tid][1][1] =  sb * (cca + ssa);
      }
    }

    #pragma unroll 1
    for (int p = 0; p < 5; ++p) {
      __syncthreads();
      // Build M16 = [[Re K, -Im K],[Im K, Re K]], K = Ua (x) Ub (x) Uc (8x8 complex)
      if (tid < 256) {
        int r = tid >> 4, c = tid & 15;
        int i = r & 7, j = c & 7;
        int wa = wtab[p][0], wb = wtab[p][1], wc = wtab[p][2];
        float ar = Ur[wa][i >> 2][j >> 2],              ai = Ui[wa][i >> 2][j >> 2];
        float br = Ur[wb][(i >> 1) & 1][(j >> 1) & 1],  bi = Ui[wb][(i >> 1) & 1][(j >> 1) & 1];
        float crr = Ur[wc][i & 1][j & 1],               cii = Ui[wc][i & 1][j & 1];
        float tr = ar * br - ai * bi;
        float ti = ar * bi + ai * br;
        float kr = tr * crr - ti * cii;
        float ki = tr * cii + ti * crr;
        M16[r][c] = ((r >> 3) == (c >> 3)) ? kr : ((r >> 3) ? ki : -ki);
      }
      __syncthreads();

      const float* reS = sRe[cur];
      const float* imS = sIm[cur];
      float* reD = sRe[cur ^ 1];
      float* imD = sIm[cur ^ 1];
      const int p0 = p0tab[p];
      if (segperm[seg] && p == 0)
        gate_pass<true >(reS, imS, reD, imD, M16, p0, lane, wave);
      else
        gate_pass<false>(reS, imS, reD, imD, M16, p0, lane, wave);
      cur ^= 1;
    }
  }

  // ---- measurement: <sum_i Z_i>, consuming the final ring permutation ----
  __syncthreads();
  const float* reS = sRe[cur];
  const float* imS = sIm[cur];
  float part = 0.f;
  for (int k = tid; k < DIM; k += NTHREADS) {
    int s = perm_inv(k);
    float re = reS[s], im = imS[s];
    part += (re * re + im * im) * (float)(NW - 2 * __popc(k));
  }
  #pragma unroll
  for (int off = 16; off > 0; off >>= 1)
    part += __shfl_down(part, off, 32);
  if (lane == 0) wsum[wave] = part;
  __syncthreads();
  if (tid == 0) {
    float t = 0.f;
    #pragma unroll
    for (int wv = 0; wv < NWAVES; ++wv) t += wsum[wv];
    out[sample] = 0.5f + 0.25f * t;           // HAM_OFFSET + HAM_COEFF * z_sum
  }
}

extern "C" void kernel_launch(void* const* d_in, const int* in_sizes, int n_in,
                              void* d_out, int out_size, void* d_ws, size_t ws_size,
                              hipStream_t stream) {
  (void)n_in; (void)out_size; (void)d_ws; (void)ws_size;
  const float* x = (const float*)d_in[0];      // (BATCH, 28) float32
  const float* w = (const float*)d_in[1];      // (4, 3, 14) float32
  float* out = (float*)d_out;                  // (BATCH, 1) float32
  int batch = in_sizes[0] / 28;
  qcirc_kernel<<<dim3(batch), dim3(NTHREADS), 0, stream>>>(x, w, out);
}